// SpargeAttention_32839319945669
// MI455X (gfx1250) — compile-verified
//
#include <hip/hip_runtime.h>
#include <hip/hip_bf16.h>

#define S_LEN 2048
#define HID 1024
#define NH 16
#define HD 64
#define NB 32
#define SCALE 0.125f    /* 1/sqrt(64) */
#define THRESH 0.01f
#define NEG_INF (-1e30f)

typedef __attribute__((ext_vector_type(2))) float v2f;
typedef __attribute__((ext_vector_type(8))) float v8f;

__device__ __forceinline__ v8f wmma4(v2f a, v2f b, v8f c) {
  // D = A(16x4 f32) x B(4x16 f32) + C(16x16 f32)
  return __builtin_amdgcn_wmma_f32_16x16x4_f32(false, a, false, b, (short)0, c,
                                               false, false);
}

__device__ __forceinline__ float redmax16(float v) {
  v = fmaxf(v, __shfl_xor(v, 1, 32));
  v = fmaxf(v, __shfl_xor(v, 2, 32));
  v = fmaxf(v, __shfl_xor(v, 4, 32));
  v = fmaxf(v, __shfl_xor(v, 8, 32));
  return v;
}
__device__ __forceinline__ float redsum16(float v) {
  v += __shfl_xor(v, 1, 32);
  v += __shfl_xor(v, 2, 32);
  v += __shfl_xor(v, 4, 32);
  v += __shfl_xor(v, 8, 32);
  return v;
}
__device__ __forceinline__ float redmax32(float v) {
  v = redmax16(v);
  v = fmaxf(v, __shfl_xor(v, 16, 32));
  return v;
}

// ---------------------------------------------------------------------------
// Y = X @ W^T.  X:[2048,1024] W:[1024,1024] (row j of W = weights of out col j)
// One wave computes a 16(M) x 64(N) tile, K-loop in steps of 4 (f32 WMMA).
// split_heads=1 -> write Y as [NH, S, HD]; else flat [S, HID].
// ---------------------------------------------------------------------------
__global__ void __launch_bounds__(32)
gemm_xwT_kernel(const float* __restrict__ X, const float* __restrict__ W,
                float* __restrict__ Y, int split_heads) {
  const int lane = threadIdx.x & 31;
  const int lr   = lane & 15;
  const int half = lane >> 4;
  const int m0   = blockIdx.x * 16;
  const int n0   = blockIdx.y * 64;

  v8f acc[4] = {};
  const float* xrow = X + (size_t)(m0 + lr) * HID + 2 * half;
  const float* wrow = W + (size_t)(n0 + lr) * HID + 2 * half;

  for (int k0 = 0; k0 < HID; k0 += 4) {
    v2f a  = *(const v2f*)(xrow + k0);
    v2f b0 = *(const v2f*)(wrow + k0);
    v2f b1 = *(const v2f*)(wrow + 16 * HID + k0);
    v2f b2 = *(const v2f*)(wrow + 32 * HID + k0);
    v2f b3 = *(const v2f*)(wrow + 48 * HID + k0);
    acc[0] = wmma4(a, b0, acc[0]);
    acc[1] = wmma4(a, b1, acc[1]);
    acc[2] = wmma4(a, b2, acc[2]);
    acc[3] = wmma4(a, b3, acc[3]);
  }

#pragma unroll
  for (int t = 0; t < 4; ++t) {
#pragma unroll
    for (int r = 0; r < 8; ++r) {
      const int row = m0 + r + 8 * half;
      const int col = n0 + 16 * t + lr;
      const float val = acc[t][r];
      if (split_heads) {
        const int h = col >> 6, d = col & 63;
        Y[((size_t)h * S_LEN + row) * HD + d] = val;
      } else {
        Y[(size_t)row * HID + col] = val;
      }
    }
  }
}

// ---------------------------------------------------------------------------
// Per-block means of q and k:  QM/KM [NH, NB, HD]
// ---------------------------------------------------------------------------
__global__ void mean_kernel(const float* __restrict__ Q,
                            const float* __restrict__ K,
                            float* __restrict__ QM, float* __restrict__ KM) {
  const int idx = blockIdx.x * blockDim.x + threadIdx.x;
  if (idx >= NH * NB * HD) return;
  const int d  = idx & (HD - 1);
  const int nb = (idx >> 6) & (NB - 1);
  const int h  = idx >> 11;
  const float* qs = Q + ((size_t)h * S_LEN + nb * 64) * HD + d;
  const float* ks = K + ((size_t)h * S_LEN + nb * 64) * HD + d;
  float sq = 0.f, sk = 0.f;
  for (int r = 0; r < 64; ++r) { sq += qs[r * HD]; sk += ks[r * HD]; }
  QM[idx] = sq * (1.0f / 64.0f);
  KM[idx] = sk * (1.0f / 64.0f);
}

// ---------------------------------------------------------------------------
// Stage-1 block mask:  (QM @ KM^T)/sqrt(D) > 0.01, one wave per 16x16 tile.
// ---------------------------------------------------------------------------
__global__ void __launch_bounds__(32)
blockscore_kernel(const float* __restrict__ QM, const float* __restrict__ KM,
                  int* __restrict__ BMask) {
  const int lane = threadIdx.x & 31;
  const int lr   = lane & 15;
  const int half = lane >> 4;
  const int m0 = blockIdx.x * 16;
  const int n0 = blockIdx.y * 16;
  const int h  = blockIdx.z;

  v8f acc = {};
  const float* qrow = QM + ((size_t)h * NB + m0 + lr) * HD + 2 * half;
  const float* krow = KM + ((size_t)h * NB + n0 + lr) * HD + 2 * half;
#pragma unroll
  for (int ks = 0; ks < 16; ++ks) {
    v2f a = *(const v2f*)(qrow + 4 * ks);
    v2f b = *(const v2f*)(krow + 4 * ks);
    acc = wmma4(a, b, acc);
  }
#pragma unroll
  for (int r = 0; r < 8; ++r) {
    const int m = m0 + r + 8 * half;
    const int n = n0 + lr;
    BMask[((size_t)h * NB + m) * NB + n] = (acc[r] * SCALE > THRESH) ? 1 : 0;
  }
}

// ---------------------------------------------------------------------------
// Sparse attention: one WG per (q-block ib, head h); 4 waves x 16 q-rows.
// Two-pass flash softmax over stage-1-allowed key blocks, then stage-2
// block-max pruning, PV via WMMA with P transposed through per-wave LDS.
// ---------------------------------------------------------------------------
__global__ void __launch_bounds__(128)
sparge_attn_kernel(const float* __restrict__ Q, const float* __restrict__ K,
                   const float* __restrict__ V, const int* __restrict__ BMask,
                   float* __restrict__ CTX) {
  __shared__ alignas(16) float lds_p[4][16 * 64];
  __shared__ float lds_bm[4];

  const int ib   = blockIdx.x;       // 64-row query block
  const int h    = blockIdx.y;
  const int w    = threadIdx.x >> 5; // wave id: rows [ib*64+16w, +16)
  const int lane = threadIdx.x & 31;
  const int lr   = lane & 15;
  const int half = lane >> 4;

  const float* Qh = Q + (size_t)h * S_LEN * HD;
  const float* Kh = K + (size_t)h * S_LEN * HD;
  const float* Vh = V + (size_t)h * S_LEN * HD;
  const int qr0 = ib * 64 + w * 16;

  // Q fragments resident for both passes (16 rows x 64 dims = 32 VGPRs)
  v2f qa[16];
  const float* qrow = Qh + (size_t)(qr0 + lr) * HD + 2 * half;
#pragma unroll
  for (int ks = 0; ks < 16; ++ks) qa[ks] = *(const v2f*)(qrow + 4 * ks);

  float mrow[8], lrow[8];
#pragma unroll
  for (int r = 0; r < 8; ++r) { mrow[r] = NEG_INF; lrow[r] = 0.0f; }

  const int* mrowp = BMask + ((size_t)h * NB + ib) * NB;

  // -------- Pass A: online row max / row sum over allowed blocks ----------
  for (int jb = 0; jb < NB; ++jb) {
    const bool m1 = (mrowp[jb] != 0);
    const bool dg = (jb == ib);
    if (!m1 && !dg) continue;

    v8f acc[4] = {};
    const float* kcol = Kh + (size_t)(jb * 64 + lr) * HD + 2 * half;
#pragma unroll 4
    for (int ks = 0; ks < 16; ++ks) {
      v2f b0 = *(const v2f*)(kcol + 4 * ks);
      v2f b1 = *(const v2f*)(kcol + 16 * HD + 4 * ks);
      v2f b2 = *(const v2f*)(kcol + 32 * HD + 4 * ks);
      v2f b3 = *(const v2f*)(kcol + 48 * HD + 4 * ks);
      acc[0] = wmma4(qa[ks], b0, acc[0]);
      acc[1] = wmma4(qa[ks], b1, acc[1]);
      acc[2] = wmma4(qa[ks], b2, acc[2]);
      acc[3] = wmma4(qa[ks], b3, acc[3]);
    }

#pragma unroll
    for (int r = 0; r < 8; ++r) {
      float sv[4];
#pragma unroll
      for (int t = 0; t < 4; ++t) {
        const float s = acc[t][r] * SCALE;
        const bool allowed = m1 || ((t == w) && (lr == r + 8 * half)); // diag
        sv[t] = allowed ? s : NEG_INF;
      }
      float mx = fmaxf(fmaxf(sv[0], sv[1]), fmaxf(sv[2], sv[3]));
      mx = redmax16(mx);
      const float mnew = fmaxf(mrow[r], mx);
      const float corr = __expf(mrow[r] - mnew);
      float ps = __expf(sv[0] - mnew) + __expf(sv[1] - mnew) +
                 __expf(sv[2] - mnew) + __expf(sv[3] - mnew);
      ps = redsum16(ps);
      lrow[r] = lrow[r] * corr + ps;
      mrow[r] = mnew;
    }
  }

  float rcl[8];
#pragma unroll
  for (int r = 0; r < 8; ++r) rcl[r] = 1.0f / lrow[r];

  // -------- Pass B: normalize, stage-2 prune, accumulate P @ V ------------
  v8f o[4] = {};
  for (int jb = 0; jb < NB; ++jb) {
    const bool m1 = (mrowp[jb] != 0);
    const bool dg = (jb == ib);
    if (!m1 && !dg) continue;

    v8f acc[4] = {};
    const float* kcol = Kh + (size_t)(jb * 64 + lr) * HD + 2 * half;
#pragma unroll 4
    for (int ks = 0; ks < 16; ++ks) {
      v2f b0 = *(const v2f*)(kcol + 4 * ks);
      v2f b1 = *(const v2f*)(kcol + 16 * HD + 4 * ks);
      v2f b2 = *(const v2f*)(kcol + 32 * HD + 4 * ks);
      v2f b3 = *(const v2f*)(kcol + 48 * HD + 4 * ks);
      acc[0] = wmma4(qa[ks], b0, acc[0]);
      acc[1] = wmma4(qa[ks], b1, acc[1]);
      acc[2] = wmma4(qa[ks], b2, acc[2]);
      acc[3] = wmma4(qa[ks], b3, acc[3]);
    }

    float p[4][8];
    float wmax = 0.0f;
#pragma unroll
    for (int t = 0; t < 4; ++t) {
#pragma unroll
      for (int r = 0; r < 8; ++r) {
        const float s = acc[t][r] * SCALE;
        const bool allowed = m1 || (dg && (t == w) && (lr == r + 8 * half));
        const float pv = allowed ? __expf(s - mrow[r]) * rcl[r] : 0.0f;
        p[t][r] = pv;
        wmax = fmaxf(wmax, pv);
      }
    }
    wmax = redmax32(wmax);

    __syncthreads();                   // protect lds_bm reuse across jb iters
    if (lane == 0) lds_bm[w] = wmax;
    __syncthreads();
    const float bm =
        fmaxf(fmaxf(lds_bm[0], lds_bm[1]), fmaxf(lds_bm[2], lds_bm[3]));
    const bool keep = bm > THRESH;

    // stage-2 masked P -> per-wave LDS (row-major 16x64) for A-layout reload
    float* pw = &lds_p[w][0];
#pragma unroll
    for (int t = 0; t < 4; ++t) {
#pragma unroll
      for (int r = 0; r < 8; ++r) {
        const bool dpos = dg && (t == w) && (lr == r + 8 * half);
        const float pv = (keep || dpos) ? p[t][r] : 0.0f;
        pw[(r + 8 * half) * 64 + 16 * t + lr] = pv;
      }
    }

    // O[16x64] += P[16x64] @ V_block[64x64]
#pragma unroll 4
    for (int ks = 0; ks < 16; ++ks) {
      v2f pa = *(const v2f*)(pw + lr * 64 + 4 * ks + 2 * half);
      const float* vrow0 = Vh + (size_t)(jb * 64 + 4 * ks + 2 * half) * HD;
      const float* vrow1 = vrow0 + HD;
#pragma unroll
      for (int t2 = 0; t2 < 4; ++t2) {
        v2f b;
        b.x = vrow0[16 * t2 + lr];
        b.y = vrow1[16 * t2 + lr];
        o[t2] = wmma4(pa, b, o[t2]);
      }
    }
  }

  // write context in [S, HID] layout for the final projection
#pragma unroll
  for (int t2 = 0; t2 < 4; ++t2) {
#pragma unroll
    for (int r = 0; r < 8; ++r) {
      const int row = qr0 + r + 8 * half;
      CTX[(size_t)row * HID + h * HD + 16 * t2 + lr] = o[t2][r];
    }
  }
}

// ---------------------------------------------------------------------------
extern "C" void kernel_launch(void* const* d_in, const int* in_sizes, int n_in,
                              void* d_out, int out_size, void* d_ws,
                              size_t ws_size, hipStream_t stream) {
  (void)in_sizes; (void)n_in; (void)out_size; (void)ws_size;
  const float* x  = (const float*)d_in[0];
  const float* wq = (const float*)d_in[1];
  const float* wk = (const float*)d_in[2];
  const float* wv = (const float*)d_in[3];
  const float* wo = (const float*)d_in[4];
  float* out = (float*)d_out;

  float* ws = (float*)d_ws;
  const size_t T = (size_t)S_LEN * HID;  // 2M floats per [S,HID]-sized tensor
  float* q   = ws;
  float* k   = q + T;
  float* v   = k + T;
  float* ctx = v + T;
  float* qm  = ctx + T;
  float* km  = qm + (size_t)NH * NB * HD;
  int*   bmask = (int*)(km + (size_t)NH * NB * HD);

  dim3 ggrid(S_LEN / 16, HID / 64);
  gemm_xwT_kernel<<<ggrid, 32, 0, stream>>>(x, wq, q, 1);
  gemm_xwT_kernel<<<ggrid, 32, 0, stream>>>(x, wk, k, 1);
  gemm_xwT_kernel<<<ggrid, 32, 0, stream>>>(x, wv, v, 1);
  mean_kernel<<<(NH * NB * HD + 255) / 256, 256, 0, stream>>>(q, k, qm, km);
  blockscore_kernel<<<dim3(NB / 16, NB / 16, NH), 32, 0, stream>>>(qm, km,
                                                                   bmask);
  sparge_attn_kernel<<<dim3(NB, NH), 128, 0, stream>>>(q, k, v, bmask, ctx);
  gemm_xwT_kernel<<<ggrid, 32, 0, stream>>>(ctx, wo, out, 0);
}